// AdaptSpikeFunction_50826642981385
// MI455X (gfx1250) — compile-verified
//
#include <hip/hip_runtime.h>
#include <stdint.h>

// Adaptive-threshold spiking scan:  z[b,t,n] = H(x[b,t,n] - thr)
//   z0 = H(x0);  t>=1: b_t = a*b_{t-1} + (1-a)*z_{t-1}*gamma_n (b seeded from b0),
//   z_t = H(x_t - b_t).
// Memory-bound: 512 MiB traffic -> ~23 us floor at 23.3 TB/s.
// Strategy: per-wave self-contained async global->LDS staging (ASYNCcnt),
// 8-stage LDS ring (128 KiB), 6 chunks in flight (~28 KiB/wave outstanding),
// ZERO barriers, non-temporal in and out.

#define ALPHA_F 0.95f
#define OMA_F   0.05f     // 1 - ALPHA

constexpr int TT  = 4096;   // timesteps (fixed by reference)
constexpr int NN  = 1024;   // neurons
constexpr int BBB = 16;     // batch

#define BLK   128           // threads per block == neurons per block (4 waves)
#define CH    32            // timesteps staged per chunk (stage = 16 KiB)
#define STG   8             // ring stages: LDS = 8*32*128*4B = 128 KiB
#define AHEAD 6             // chunks issued ahead: max 56 outstanding asyncs <= 63
constexpr int NC   = TT / CH;    // 128 chunks
constexpr int NBLK = NN / BLK;   // 8 neuron slabs per batch

__global__ __launch_bounds__(BLK)
void adapt_spike_scan_kernel(const float* __restrict__ x,
                             const float* __restrict__ gamma,
                             const float* __restrict__ b0,
                             float* __restrict__ out)
{
    __shared__ float sbuf[STG * CH * BLK];

    const int tid = threadIdx.x;
    const int bb  = blockIdx.x >> 3;                 // / NBLK
    const int n0  = (blockIdx.x & (NBLK - 1)) * BLK; // first neuron of slab
    const int n   = n0 + tid;

    const size_t rowbase = ((size_t)bb * TT) * NN + (size_t)n0;
    const unsigned lds_base = (unsigned)(uintptr_t)(&sbuf[0]);

    // Wave-local staging: wave w stages exactly the 32-neuron strip its own
    // lanes consume -> ASYNCcnt (per-wave, in-order) is the only sync needed.
    // Lane l of wave w stages float4s at rows (q*4 + l>>3), cols 32w + 4*(l&7).
    const int w    = tid >> 5;                       // wave in block
    const int l    = tid & 31;                       // lane in wave
    const int srow = l >> 3;                         // 0..3
    const int scol = (w << 5) + ((l & 7) << 2);      // float column of lane's b128

    auto issue = [&](int c, int s) {
        uint64_t ga = (uint64_t)(uintptr_t)
            (x + rowbase + (size_t)(c * CH + srow) * NN + (size_t)scol);
        unsigned la = lds_base + (unsigned)(((s * CH + srow) * BLK + scol) * 4);
        #pragma unroll
        for (int q = 0; q < CH / 4; ++q) {           // 8 async b128 per wave/chunk
            asm volatile("global_load_async_to_lds_b128 %0, %1, off th:TH_LOAD_NT"
                         :: "v"(la), "v"(ga) : "memory");
            ga += (uint64_t)(4 * NN * 4);            // 4 rows forward in global
            la += 4 * BLK * 4;                       // 4 rows forward in LDS
        }
    };

    const float g = gamma[n];
    float bs      = b0[n];
    float zprev   = 0.0f;

    auto compute = [&](int c) {
        const float* buf = &sbuf[(c & (STG - 1)) * CH * BLK];
        const size_t obase = rowbase + (size_t)(c * CH) * NN + (size_t)tid;
        #pragma unroll
        for (int k = 0; k < CH; ++k) {
            const float xv = buf[k * BLK + tid];     // ds_load, conflict-free
            bs = ALPHA_F * bs + OMA_F * zprev * g;
            const float z = ((xv - bs) > 0.0f) ? 1.0f : 0.0f;
            zprev = z;
            __builtin_nontemporal_store(z, &out[obase + (size_t)k * NN]);
        }
    };

    // Prologue: fill the pipeline AHEAD chunks deep.
    #pragma unroll
    for (int c = 0; c < AHEAD; ++c) issue(c, c);

    // ---- chunk 0: t==0 is a raw Heaviside; recurrence starts at k==1 ----
    issue(AHEAD, AHEAD);
    asm volatile("s_wait_asynccnt 48" ::: "memory"); // this wave's chunk 0 landed
    {
        const float* buf = &sbuf[0];
        float xv = buf[tid];
        float z  = (xv > 0.0f) ? 1.0f : 0.0f;
        zprev = z;
        __builtin_nontemporal_store(z, &out[rowbase + (size_t)tid]);
        #pragma unroll
        for (int k = 1; k < CH; ++k) {
            xv = buf[k * BLK + tid];
            bs = ALPHA_F * bs + OMA_F * zprev * g;
            z  = ((xv - bs) > 0.0f) ? 1.0f : 0.0f;
            zprev = z;
            __builtin_nontemporal_store(z, &out[rowbase + (size_t)k * NN + (size_t)tid]);
        }
    }

    // ---- steady state: constant wait, no branching on the wait ----
    for (int c = 1; c < NC - AHEAD; ++c) {
        issue(c + AHEAD, (c + AHEAD) & (STG - 1));
        asm volatile("s_wait_asynccnt 48" ::: "memory");  // chunk c complete
        compute(c);
    }

    // ---- drain the last AHEAD chunks ----
    for (int c = NC - AHEAD; c < NC; ++c) {
        const int rem = NC - 1 - c;                  // 5..0 chunks still newer
        if      (rem == 5) asm volatile("s_wait_asynccnt 40" ::: "memory");
        else if (rem == 4) asm volatile("s_wait_asynccnt 32" ::: "memory");
        else if (rem == 3) asm volatile("s_wait_asynccnt 24" ::: "memory");
        else if (rem == 2) asm volatile("s_wait_asynccnt 16" ::: "memory");
        else if (rem == 1) asm volatile("s_wait_asynccnt 8"  ::: "memory");
        else               asm volatile("s_wait_asynccnt 0"  ::: "memory");
        compute(c);
    }
}

extern "C" void kernel_launch(void* const* d_in, const int* in_sizes, int n_in,
                              void* d_out, int out_size, void* d_ws, size_t ws_size,
                              hipStream_t stream) {
    const float* x     = (const float*)d_in[0];   // [16, 4096, 1024] f32
    const float* gamma = (const float*)d_in[1];   // [1024] f32
    const float* b0    = (const float*)d_in[2];   // [1024] f32
    float* out         = (float*)d_out;           // [16, 4096, 1024] f32

    dim3 grid(BBB * NBLK);                        // 128 blocks
    dim3 block(BLK);                              // 128 threads (4 waves)
    hipLaunchKernelGGL(adapt_spike_scan_kernel, grid, block, 0, stream,
                       x, gamma, b0, out);
}